// KVCache_21388937134425
// MI455X (gfx1250) — compile-verified
//
#include <hip/hip_runtime.h>
#include <math.h>

#define WINDOW_  64
#define BUDGET_  256
#define MAXGEN_  128
#define B_       4
#define H_       32
#define HK_      8
#define REP_     4
#define D_       128
#define T_       (BUDGET_ + WINDOW_ + MAXGEN_)   // 448
#define CLMAX_   4096
#define KSTRIDE_ 130                              // padded LDS row stride

typedef float v2f __attribute__((ext_vector_type(2)));
typedef float v8f __attribute__((ext_vector_type(8)));

__device__ __forceinline__ v8f wmma_f32_16x16x4(v2f a, v2f b, v8f c) {
  // D = A(16x4 f32) * B(4x16 f32) + C(16x16 f32)
  return __builtin_amdgcn_wmma_f32_16x16x4_f32(
      /*neg_a=*/false, a, /*neg_b=*/false, b,
      /*c_mod=*/(short)0, c, /*reuse_a=*/false, /*reuse_b=*/false);
}

// Async global->LDS copy, 16 bytes per lane.  Tracked by ASYNCcnt.
// LDS byte address = low 32 bits of the generic shared pointer
// (ISA 10.2: LDS aperture truncates flat address to addr[31:0]).
__device__ __forceinline__ void async_copy_b128(void* lds_dst, const void* gsrc) {
  unsigned l = (unsigned)(size_t)lds_dst;
  asm volatile("global_load_async_to_lds_b128 %0, %1, off"
               :: "v"(l), "v"(gsrc) : "memory");
}

__device__ __forceinline__ void wait_async0() {
#if __has_builtin(__builtin_amdgcn_s_wait_asynccnt)
  __builtin_amdgcn_s_wait_asynccnt(0);
#else
  asm volatile("s_wait_asynccnt 0x0" ::: "memory");
#endif
}

// ---------------------------------------------------------------------------
// Kernel 1: attention vote.  One block per (b, hk, rh); 128 threads = 4 waves.
// Wave w owns query rows l in [w*16, w*16+16) for head h = hk*REP_+rh.
// K streamed through a double-buffered LDS tile (32 keys x 128 dims) filled
// with global_load_async_to_lds_b128 and gated by s_wait_asynccnt.
// vote workspace layout: [(b*HK_+hk)*REP_+rh][CLMAX_]  (each block owns slice)
// ---------------------------------------------------------------------------
__global__ __launch_bounds__(128)
void vote_kernel(const float* __restrict__ q,
                 const float* __restrict__ k_cache,
                 const int*   __restrict__ cache_len_ptr,
                 float*       __restrict__ vote,
                 int Smax)
{
  const int CL = min(*cache_len_ptr, CLMAX_);
  const int S  = CL + WINDOW_;
  const int nchunks  = S  >> 5;   // 32-key chunks, pass 1 (stats incl. window)
  const int nchunksv = CL >> 5;   // pass 2 (vote: only s < cache_len)

  const int bid = blockIdx.x;            // b*HK_*REP_ + hk*REP_ + rh
  const int rh  = bid % REP_;
  const int hk  = (bid / REP_) % HK_;
  const int b   = bid / (REP_ * HK_);
  const int h   = hk * REP_ + rh;

  const int tid  = threadIdx.x;
  const int lane = tid & 31;
  const int wave = tid >> 5;             // 0..3

  __shared__ float sK[2][32 * KSTRIDE_]; // double-buffered 32 keys x 128 dims
  __shared__ float sVote[CLMAX_];
  __shared__ float sCol[4][32];          // per-wave column sums (deterministic)

  for (int i = tid; i < CLMAX_; i += 128) sVote[i] = 0.f;

  // ---- A fragments: q rows for this wave, f32 WMMA 16x4 layout ----
  // lanes 0-15: K = 4*kc + {0,1}; lanes 16-31: K = 4*kc + {2,3}; row = lane&15
  const int arow = wave * 16 + (lane & 15);
  const int koff = (lane & 16) ? 2 : 0;
  v2f qa[32];
  {
    const float* qrow = q + (((size_t)b * WINDOW_ + arow) * H_ + h) * D_;
    #pragma unroll
    for (int kc = 0; kc < 32; ++kc) {
      float2 t2 = *reinterpret_cast<const float2*>(qrow + 4 * kc + koff);
      v2f a = {t2.x, t2.y};
      qa[kc] = a;
    }
  }

  // async fill of one 32-key chunk: 1024 float4 slots, 8 per thread
  auto issue_chunk = [&](int c, int buf) {
    const int s0 = c << 5;
    #pragma unroll
    for (int j = 0; j < 8; ++j) {
      const int f   = j * 128 + tid;       // float4 slot in tile
      const int key = f >> 5;              // 32 float4 per key row
      const int d4  = (f & 31) << 2;
      const float* g =
          k_cache + (((size_t)b * Smax + (s0 + key)) * HK_ + hk) * D_ + d4;
      async_copy_b128(&sK[buf][key * KSTRIDE_ + d4], g);
    }
  };

  const float scale = 0.08838834764831845f;   // 1/sqrt(128)
  const int   bcol  = lane & 15;              // B-frag N / C col
  const int   crow  = (lane & 16) ? 8 : 0;    // C-layout row offset per half

  float rm[8], rl[8];
  #pragma unroll
  for (int r = 0; r < 8; ++r) { rm[r] = -__builtin_inff(); rl[r] = 0.f; }

  // ================= pass 1: online softmax stats =================
  issue_chunk(0, 0);
  for (int c = 0; c < nchunks; ++c) {
    const int buf = c & 1;
    const int s0  = c << 5;
    wait_async0();            // my copies for chunk c are in LDS
    __syncthreads();          // everyone's copies done; buf^1 free to refill
    if (c + 1 < nchunks) issue_chunk(c + 1, buf ^ 1);

    const float* kb = &sK[buf][0];
    v8f acc0 = {}, acc1 = {};
    #pragma unroll
    for (int kc = 0; kc < 32; ++kc) {
      float2 t0 = *reinterpret_cast<const float2*>(
          &kb[bcol * KSTRIDE_ + 4 * kc + koff]);
      float2 t1 = *reinterpret_cast<const float2*>(
          &kb[(bcol + 16) * KSTRIDE_ + 4 * kc + koff]);
      v2f b0 = {t0.x, t0.y};
      v2f b1 = {t1.x, t1.y};
      acc0 = wmma_f32_16x16x4(qa[kc], b0, acc0);
      acc1 = wmma_f32_16x16x4(qa[kc], b1, acc1);
    }

    const int sA = s0 + bcol;        // key index, first 16-col tile
    const int sB = s0 + 16 + bcol;   // key index, second tile
    #pragma unroll
    for (int r = 0; r < 8; ++r) {
      const int l = wave * 16 + r + crow;       // query index of this value
      float sc0 = acc0[r] * scale;
      float sc1 = acc1[r] * scale;
      if (sA >= CL && (sA - CL) > l) sc0 = -__builtin_inff();  // causal mask
      if (sB >= CL && (sB - CL) > l) sc1 = -__builtin_inff();
      // row max over both tiles, within each 16-lane half
      float v = fmaxf(sc0, sc1);
      v = fmaxf(v, __shfl_xor(v, 1, 32));
      v = fmaxf(v, __shfl_xor(v, 2, 32));
      v = fmaxf(v, __shfl_xor(v, 4, 32));
      v = fmaxf(v, __shfl_xor(v, 8, 32));
      const float newm = fmaxf(rm[r], v);
      float e = ((sc0 == -__builtin_inff()) ? 0.f : __expf(sc0 - newm)) +
                ((sc1 == -__builtin_inff()) ? 0.f : __expf(sc1 - newm));
      e += __shfl_xor(e, 1, 32);
      e += __shfl_xor(e, 2, 32);
      e += __shfl_xor(e, 4, 32);
      e += __shfl_xor(e, 8, 32);
      const float corr =
          (rm[r] == -__builtin_inff()) ? 0.f : __expf(rm[r] - newm);
      rl[r] = rl[r] * corr + e;
      rm[r] = newm;
    }
  }

  // ================= pass 2: vote accumulation (s < CL) =================
  __syncthreads();            // nobody still reading pass-1 buffers
  issue_chunk(0, 0);
  for (int c = 0; c < nchunksv; ++c) {
    const int buf = c & 1;
    const int s0  = c << 5;
    wait_async0();
    __syncthreads();
    if (c + 1 < nchunksv) issue_chunk(c + 1, buf ^ 1);

    const float* kb = &sK[buf][0];
    v8f acc0 = {}, acc1 = {};
    #pragma unroll
    for (int kc = 0; kc < 32; ++kc) {
      float2 t0 = *reinterpret_cast<const float2*>(
          &kb[bcol * KSTRIDE_ + 4 * kc + koff]);
      float2 t1 = *reinterpret_cast<const float2*>(
          &kb[(bcol + 16) * KSTRIDE_ + 4 * kc + koff]);
      v2f b0 = {t0.x, t0.y};
      v2f b1 = {t1.x, t1.y};
      acc0 = wmma_f32_16x16x4(qa[kc], b0, acc0);
      acc1 = wmma_f32_16x16x4(qa[kc], b1, acc1);
    }

    float cs0 = 0.f, cs1 = 0.f;
    #pragma unroll
    for (int r = 0; r < 8; ++r) {
      cs0 += __expf(acc0[r] * scale - rm[r]) / rl[r];
      cs1 += __expf(acc1[r] * scale - rm[r]) / rl[r];
    }
    cs0 += __shfl_xor(cs0, 16, 32);   // + other half => full 16-row col sums
    cs1 += __shfl_xor(cs1, 16, 32);
    if (lane < 16) {                  // stage per wave (deterministic reduce)
      sCol[wave][bcol]      = cs0;
      sCol[wave][bcol + 16] = cs1;
    }
    __syncthreads();
    if (tid < 32)
      sVote[s0 + tid] +=
          sCol[0][tid] + sCol[1][tid] + sCol[2][tid] + sCol[3][tid];
  }

  __syncthreads();
  float* vslice = vote + (size_t)bid * CLMAX_;
  for (int i = tid; i < CLMAX_; i += 128) vslice[i] = sVote[i];
}

// ---------------------------------------------------------------------------
// Kernel 2: box-pool + stable top-256 + gather.  One block per (b, hk).
// ---------------------------------------------------------------------------
__global__ __launch_bounds__(256)
void select_gather_kernel(const float* __restrict__ k_cache,
                          const float* __restrict__ v_cache,
                          const int*   __restrict__ cache_len_ptr,
                          const float* __restrict__ vote,
                          float*       __restrict__ out,
                          int Smax)
{
  const int CL  = min(*cache_len_ptr, CLMAX_);
  const int bid = blockIdx.x;               // b*HK_ + hk
  const int hk  = bid % HK_;
  const int b   = bid / HK_;
  const int tid = threadIdx.x;

  __shared__ float sVal[CLMAX_];
  __shared__ float sPool[CLMAX_];
  __shared__ float sRedV[256];
  __shared__ int   sRedI[256];
  __shared__ int   sTop[BUDGET_];

  // sum the 4 rep slices (deterministic order)
  const float* vbase = vote + (size_t)bid * REP_ * CLMAX_;
  for (int i = tid; i < CLMAX_; i += 256) {
    float s = 0.f;
    #pragma unroll
    for (int rr = 0; rr < REP_; ++rr) s += vbase[(size_t)rr * CLMAX_ + i];
    sVal[i] = (i < CL) ? s : 0.f;
  }
  __syncthreads();

  // zero-padded box filter, width 31 (÷31 omitted: monotonic for top-k)
  for (int i = tid; i < CLMAX_; i += 256) {
    if (i < CL) {
      const int lo = max(i - 15, 0), hi = min(i + 15, CL - 1);
      float s = 0.f;
      for (int j = lo; j <= hi; ++j) s += sVal[j];
      sPool[i] = s;
    } else {
      sPool[i] = -__builtin_inff();
    }
  }
  __syncthreads();

  // stable top-256: repeated argmax, ties -> lowest index (lax.top_k order)
  for (int r = 0; r < BUDGET_; ++r) {
    float bv = -__builtin_inff();
    int   bi = 0x7fffffff;
    for (int i = tid; i < CLMAX_; i += 256) {
      const float v = sPool[i];
      if (v > bv || (v == bv && i < bi)) { bv = v; bi = i; }
    }
    sRedV[tid] = bv; sRedI[tid] = bi;
    __syncthreads();
    for (int off = 128; off > 0; off >>= 1) {
      if (tid < off) {
        const float ov = sRedV[tid + off];
        const int   oi = sRedI[tid + off];
        if (ov > sRedV[tid] || (ov == sRedV[tid] && oi < sRedI[tid])) {
          sRedV[tid] = ov; sRedI[tid] = oi;
        }
      }
      __syncthreads();
    }
    if (tid == 0) { sTop[r] = sRedI[0]; sPool[sRedI[0]] = -__builtin_inff(); }
    __syncthreads();
  }

  // gather: out shape (2, B, T, Hk, D)
  const size_t plane = (size_t)B_ * T_ * HK_ * D_;
  float* outK = out;
  float* outV = out + plane;

  for (int u = tid; u < BUDGET_ * (D_ / 4); u += 256) {
    const int i  = u >> 5;               // D_/4 == 32
    const int d4 = (u & 31) << 2;
    const int s  = sTop[i];
    const size_t src = (((size_t)b * Smax + s) * HK_ + hk) * D_ + d4;
    const size_t dst = (((size_t)b * T_  + i) * HK_ + hk) * D_ + d4;
    *reinterpret_cast<float4*>(outK + dst) =
        *reinterpret_cast<const float4*>(k_cache + src);
    *reinterpret_cast<float4*>(outV + dst) =
        *reinterpret_cast<const float4*>(v_cache + src);
  }
  for (int u = tid; u < WINDOW_ * (D_ / 4); u += 256) {
    const int i  = u >> 5;
    const int d4 = (u & 31) << 2;
    const size_t src = (((size_t)b * Smax + (CL + i)) * HK_ + hk) * D_ + d4;
    const size_t dst = (((size_t)b * T_ + (BUDGET_ + i)) * HK_ + hk) * D_ + d4;
    *reinterpret_cast<float4*>(outK + dst) =
        *reinterpret_cast<const float4*>(k_cache + src);
    *reinterpret_cast<float4*>(outV + dst) =
        *reinterpret_cast<const float4*>(v_cache + src);
  }
  for (int u = tid; u < MAXGEN_ * (D_ / 4); u += 256) {
    const int i  = u >> 5;
    const int d4 = (u & 31) << 2;
    const size_t dst =
        (((size_t)b * T_ + (BUDGET_ + WINDOW_ + i)) * HK_ + hk) * D_ + d4;
    const float4 z = {0.f, 0.f, 0.f, 0.f};
    *reinterpret_cast<float4*>(outK + dst) = z;
    *reinterpret_cast<float4*>(outV + dst) = z;
  }
}

// ---------------------------------------------------------------------------
extern "C" void kernel_launch(void* const* d_in, const int* in_sizes, int n_in,
                              void* d_out, int out_size, void* d_ws, size_t ws_size,
                              hipStream_t stream)
{
  const float* q         = (const float*)d_in[0];
  const float* k_cache   = (const float*)d_in[1];
  const float* v_cache   = (const float*)d_in[2];
  const int*   cache_len = (const int*)d_in[3];

  const int Smax = in_sizes[1] / (B_ * HK_ * D_);   // 8192

  float* vote = (float*)d_ws;  // B_*HK_*REP_*CLMAX_ floats = 2 MiB, fully
                               // overwritten by vote_kernel (no memset needed)

  vote_kernel<<<B_ * HK_ * REP_, 128, 0, stream>>>(
      q, k_cache, cache_len, vote, Smax);
  select_gather_kernel<<<B_ * HK_, 256, 0, stream>>>(
      k_cache, v_cache, cache_len, vote, (float*)d_out, Smax);
}